// GNNlayer_20761871909292
// MI455X (gfx1250) — compile-verified
//
#include <hip/hip_runtime.h>

// Problem constants (from reference): B=2, CIN=128, COUT=64, K=64, H=W=128
#define HW    16384     // H*W = 2^14
#define BB    2
#define CIN_  128
#define COUT_ 64
#define KK    64
#define BN_EPS 1e-5f

typedef __attribute__((ext_vector_type(2))) float v2f;
typedef __attribute__((ext_vector_type(8))) float v8f;

// ---- workspace layout (in floats) ----
#define XT_F      0                      // xt/features [B][HW][COUT]  : 2*16384*64 = 2097152
#define SUMS_F    2097152                // sums [B][K][COUT]          : 8192
#define CNT_F     (SUMS_F + 8192)        // counts [B][K]              : 128
#define INVC_F    (CNT_F + 128)          // inv_cov [64][64]           : 4096
#define ADJM_F    (INVC_F + 4096)        // adj_means [B][K][COUT]     : 8192
#define BNS_F     (ADJM_F + 8192)        // bn sum  [COUT]             : 64
#define BNQ_F     (BNS_F + 64)           // bn sumsq[COUT]             : 64
#define WS_TOTAL_F (BNQ_F + 64)
#define ACC_ZERO_N (WS_TOTAL_F - SUMS_F) // accumulators that must start at 0 each call

// ---------------------------------------------------------------------------
// 0) zero all accumulator regions (re-run every call: graph replay safe)
// ---------------------------------------------------------------------------
__global__ void init_ws_kernel(float* __restrict__ ws) {
  int t = blockIdx.x * blockDim.x + threadIdx.x;
  if (t < ACC_ZERO_N) ws[SUMS_F + t] = 0.0f;
}

// ---------------------------------------------------------------------------
// 1) xt[b,p,o] = sum_c x[b,c,p] * Wft[c,o]   via V_WMMA_F32_16X16X4_F32
//    One wave => 16-pixel x 64-channel tile, K accumulated in steps of 4.
//    A frag (16x4 f32): lane L<16 holds M=L, K={k0,k0+1}; L>=16 holds K={k0+2,k0+3}
//    B frag (4x16 f32): same K split across lane halves, N = lane%16
//    D (16x16 f32, 8 VGPRs): M = r + 8*(lane>=16), N = lane%16
// ---------------------------------------------------------------------------
__global__ __launch_bounds__(128) void xt_wmma_kernel(const float* __restrict__ x,
                                                      const float* __restrict__ Wft,
                                                      float* __restrict__ xt) {
  const int lane   = threadIdx.x & 31;
  const int wid    = threadIdx.x >> 5;
  const int tile   = blockIdx.x * 4 + wid;        // 2048 tiles total
  const int tilesPerB = HW / 16;                  // 1024
  const int b      = tile / tilesPerB;
  const int pbase  = (tile % tilesPerB) * 16;
  const int lanelo = lane & 15;
  const int lanehi = lane >> 4;                   // 0 or 1

  v8f acc[4];
  #pragma unroll
  for (int n = 0; n < 4; ++n) acc[n] = (v8f){0.f,0.f,0.f,0.f,0.f,0.f,0.f,0.f};

  #pragma unroll 4
  for (int k0 = 0; k0 < CIN_; k0 += 4) {
    const int c0 = k0 + 2 * lanehi;
    // A fragment: x[b, c0..c0+1, pbase+lanelo]  (pixels contiguous -> coalesced)
    const float* xp = x + ((size_t)b * CIN_ + c0) * HW + (pbase + lanelo);
    v2f a;
    a.x = xp[0];
    a.y = xp[HW];

    #pragma unroll
    for (int n = 0; n < 4; ++n) {
      const float* wp = Wft + (size_t)c0 * COUT_ + n * 16 + lanelo;
      v2f bf;
      bf.x = wp[0];
      bf.y = wp[COUT_];
      acc[n] = __builtin_amdgcn_wmma_f32_16x16x4_f32(
          /*neg_a=*/false, a, /*neg_b=*/false, bf,
          /*c_mod=*/(short)0, acc[n], /*reuse_a=*/false, /*reuse_b=*/false);
    }
  }

  // store pixel-major: xt[(b*HW + p)*COUT + o]  (16 lanes write 64B runs)
  #pragma unroll
  for (int n = 0; n < 4; ++n) {
    #pragma unroll
    for (int r = 0; r < 8; ++r) {
      const int p = pbase + r + 8 * lanehi;
      xt[((size_t)b * HW + p) * COUT_ + n * 16 + lanelo] = acc[n][r];
    }
  }
}

// ---------------------------------------------------------------------------
// 2) scatter: sums[b,k,o] += xt[b,p,o], counts[b,k] += 1  (LDS-aggregated)
//    64 threads cooperate per pixel (o = tid&63) -> conflict-free ds_add_f32
// ---------------------------------------------------------------------------
__global__ __launch_bounds__(256) void scatter_kernel(const float* __restrict__ xt,
                                                      const int* __restrict__ index,
                                                      float* __restrict__ sums,
                                                      float* __restrict__ counts) {
  __shared__ float lsum[KK * COUT_];
  __shared__ float lcnt[KK];
  const int tid = threadIdx.x;
  for (int e = tid; e < KK * COUT_; e += 256) lsum[e] = 0.0f;
  if (tid < KK) lcnt[tid] = 0.0f;
  __syncthreads();

  const int o   = tid & 63;
  const int grp = tid >> 6;                 // 0..3
  const int pixbase = blockIdx.x * 1024;    // block fully inside one batch
  const int b = pixbase / HW;

  for (int lp = grp; lp < 1024; lp += 4) {
    const int g = pixbase + lp;             // flat pixel over B*HW
    const int k = index[g];
    atomicAdd(&lsum[k * COUT_ + o], xt[(size_t)g * COUT_ + o]);
    if (o == 0) atomicAdd(&lcnt[k], 1.0f);
  }
  __syncthreads();

  for (int e = tid; e < KK * COUT_; e += 256)
    atomicAdd(&sums[(size_t)b * KK * COUT_ + e], lsum[e]);
  if (tid < KK) atomicAdd(&counts[b * KK + tid], lcnt[tid]);
}

// ---------------------------------------------------------------------------
// 3) cov = Wm Wm^T (SPD); invert via Gauss-Jordan (no pivoting) in LDS
// ---------------------------------------------------------------------------
__global__ __launch_bounds__(256) void covinv_kernel(const float* __restrict__ Wm,
                                                     float* __restrict__ invC) {
  __shared__ float A[64 * 64];
  __shared__ float Inv[64 * 64];
  __shared__ float colf[64];
  const int tid = threadIdx.x;

  for (int e = tid; e < 4096; e += 256) {
    const int i = e >> 6, j = e & 63;
    float s = 0.0f;
    for (int c = 0; c < 64; ++c) s += Wm[i * 64 + c] * Wm[j * 64 + c];
    A[e]   = s;
    Inv[e] = (i == j) ? 1.0f : 0.0f;
  }
  __syncthreads();

  for (int k = 0; k < 64; ++k) {
    const float pinv = 1.0f / A[k * 64 + k];
    if (tid >= 128 && tid < 192) {
      const int i = tid - 128;
      colf[i] = (i == k) ? 0.0f : A[i * 64 + k];
    }
    __syncthreads();
    if (tid < 64)        A[k * 64 + tid]        *= pinv;
    else if (tid < 128)  Inv[k * 64 + tid - 64] *= pinv;
    __syncthreads();
    for (int e = tid; e < 4096; e += 256) {
      const int i = e >> 6, j = e & 63;
      if (i != k) {
        const float f = colf[i];
        A[e]   -= f * A[k * 64 + j];
        Inv[e] -= f * Inv[k * 64 + j];
      }
    }
    __syncthreads();
  }
  for (int e = tid; e < 4096; e += 256) invC[e] = Inv[e];
}

// ---------------------------------------------------------------------------
// 4) per batch: means, Mahalanobis adjacency, adj_means (all in LDS)
//    q[i,j] = s_i + s_j - 2*G[i,j],  G = M invC M^T,  s_i = G[i,i]
// ---------------------------------------------------------------------------
__global__ __launch_bounds__(256) void adj_kernel(const float* __restrict__ sums,
                                                  const float* __restrict__ counts,
                                                  const float* __restrict__ invC,
                                                  float* __restrict__ adjm) {
  __shared__ float sm[64 * 64];   // means
  __shared__ float su[64 * 64];   // u = means @ invC, then adj
  __shared__ float sc[64 * 64];   // invC
  __shared__ float sv[64];        // s_i
  const int tid = threadIdx.x;
  const int b = blockIdx.x;

  for (int e = tid; e < 4096; e += 256) {
    const int k = e >> 6;
    const float cnt = counts[b * KK + k];
    const float denom = cnt + (cnt == 0.0f ? 1.0f : 0.0f);
    sm[e] = sums[(size_t)b * 4096 + e] / denom;
    sc[e] = invC[e];
  }
  __syncthreads();

  for (int e = tid; e < 4096; e += 256) {   // u[i,c] = sum_d m[i,d] invC[d,c]
    const int i = e >> 6, c = e & 63;
    float s = 0.0f;
    for (int d = 0; d < 64; ++d) s += sm[i * 64 + d] * sc[d * 64 + c];
    su[e] = s;
  }
  __syncthreads();

  if (tid < 64) {
    float s = 0.0f;
    for (int c = 0; c < 64; ++c) s += sm[tid * 64 + c] * su[tid * 64 + c];
    sv[tid] = s;
  }
  __syncthreads();

  float g[16];
  #pragma unroll
  for (int t = 0; t < 16; ++t) {
    const int e = t * 256 + tid;
    const int i = e >> 6, j = e & 63;
    float G = 0.0f;
    for (int c = 0; c < 64; ++c) G += sm[i * 64 + c] * su[j * 64 + c];
    float q = sv[i] + sv[j] - 2.0f * G;
    q = fmaxf(q, 1e-12f);
    g[t] = expf(-sqrtf(q));
  }
  __syncthreads();
  #pragma unroll
  for (int t = 0; t < 16; ++t) su[t * 256 + tid] = g[t];   // su := adj
  __syncthreads();

  for (int e = tid; e < 4096; e += 256) {   // adj_means[i,o] = sum_j adj[i,j] m[j,o]
    const int i = e >> 6, o = e & 63;
    float s = 0.0f;
    for (int j = 0; j < 64; ++j) s += su[i * 64 + j] * sm[j * 64 + o];
    adjm[(size_t)b * 4096 + e] = s;
  }
}

// ---------------------------------------------------------------------------
// 5) features = relu(xt + adj_means[idx]) (in place) + per-channel BN stats
// ---------------------------------------------------------------------------
__global__ __launch_bounds__(256) void feat_kernel(float* __restrict__ xt,
                                                   const int* __restrict__ index,
                                                   const float* __restrict__ adjm,
                                                   float* __restrict__ bns,
                                                   float* __restrict__ bnq) {
  __shared__ float ls[COUT_];
  __shared__ float lq[COUT_];
  const int tid = threadIdx.x;
  if (tid < COUT_) { ls[tid] = 0.0f; lq[tid] = 0.0f; }
  __syncthreads();

  const int o  = tid & 63;
  const int pr = tid >> 6;
  const int pixbase = blockIdx.x * 64;    // 64 pixels per block, same batch
  const int b = pixbase / HW;

  float s = 0.0f, s2 = 0.0f;
  for (int lp = pr; lp < 64; lp += 4) {
    const int g = pixbase + lp;
    const int k = index[g];
    const float ga = adjm[((size_t)b * KK + k) * COUT_ + o];
    float f = xt[(size_t)g * COUT_ + o] + ga;
    f = fmaxf(f, 0.0f);
    xt[(size_t)g * COUT_ + o] = f;
    s += f;
    s2 += f * f;
  }
  atomicAdd(&ls[o], s);
  atomicAdd(&lq[o], s2);
  __syncthreads();
  if (tid < COUT_) { atomicAdd(&bns[tid], ls[tid]); atomicAdd(&bnq[tid], lq[tid]); }
}

// ---------------------------------------------------------------------------
// 6) BN finalize + transpose to (B, COUT, H, W) output layout
// ---------------------------------------------------------------------------
__global__ __launch_bounds__(256) void bn_out_kernel(const float* __restrict__ xt,
                                                     const float* __restrict__ bns,
                                                     const float* __restrict__ bnq,
                                                     const float* __restrict__ gamma,
                                                     const float* __restrict__ beta,
                                                     float* __restrict__ out) {
  const int t = blockIdx.x * 256 + threadIdx.x;     // over B*COUT*HW = 2^21
  const int p = t & (HW - 1);
  const int o = (t >> 14) & 63;
  const int b = t >> 20;
  const float invN = 1.0f / (float)(BB * HW);
  const float m = bns[o] * invN;
  const float v = bnq[o] * invN - m * m;
  const float inv = rsqrtf(v + BN_EPS);
  const float f = xt[((size_t)b * HW + p) * COUT_ + o];
  out[t] = (f - m) * inv * gamma[o] + beta[o];
}

// ---------------------------------------------------------------------------
extern "C" void kernel_launch(void* const* d_in, const int* in_sizes, int n_in,
                              void* d_out, int out_size, void* d_ws, size_t ws_size,
                              hipStream_t stream) {
  const float* x     = (const float*)d_in[0];   // (B,CIN,H,W) f32
  const int*   index = (const int*)  d_in[1];   // (B,1,H,W)   i32
  const float* Wft   = (const float*)d_in[2];   // (CIN,COUT)  f32
  const float* Wm    = (const float*)d_in[3];   // (COUT,COUT) f32
  const float* gamma = (const float*)d_in[4];   // (COUT,)     f32
  const float* beta  = (const float*)d_in[5];   // (COUT,)     f32
  float* out = (float*)d_out;
  float* ws  = (float*)d_ws;

  float* xt   = ws + XT_F;
  float* sums = ws + SUMS_F;
  float* cnts = ws + CNT_F;
  float* invC = ws + INVC_F;
  float* adjm = ws + ADJM_F;
  float* bns  = ws + BNS_F;
  float* bnq  = ws + BNQ_F;

  init_ws_kernel<<<(ACC_ZERO_N + 255) / 256, 256, 0, stream>>>(ws);
  xt_wmma_kernel<<<512, 128, 0, stream>>>(x, Wft, xt);
  scatter_kernel<<<(BB * HW) / 1024, 256, 0, stream>>>(xt, index, sums, cnts);
  covinv_kernel<<<1, 256, 0, stream>>>(Wm, invC);
  adj_kernel<<<BB, 256, 0, stream>>>(sums, cnts, invC, adjm);
  feat_kernel<<<(BB * HW) / 64, 256, 0, stream>>>(xt, index, adjm, bns, bnq);
  bn_out_kernel<<<(BB * COUT_ * HW) / 256, 256, 0, stream>>>(xt, bns, bnq, gamma, beta, out);
}